// TGCN_42631845380487
// MI455X (gfx1250) — compile-verified
//
#include <hip/hip_runtime.h>
#include <hip/hip_bf16.h>
#include <math.h>

// ---------------------------------------------------------------------------
// GNN TransformerConv x3 for MI455X (gfx1250, wave32).
// Dense node projections -> V_WMMA_F32_16X16X4_F32 (fp32 matrix pipe),
// weights staged to LDS with GLOBAL_LOAD_ASYNC_TO_LDS_B128 (+ s_wait_asynccnt).
// Edge phase -> VALU + global float atomics (memory-bound scatter/gather),
// with the edge-projection matrix We broadcast from LDS.
// ---------------------------------------------------------------------------

typedef __attribute__((ext_vector_type(2))) float v2f;
typedef __attribute__((ext_vector_type(8))) float v8f;

#define N_NODES 100000
#define N_EDGES 1600000

__device__ __forceinline__ void wait_async0() {
#if __has_builtin(__builtin_amdgcn_s_wait_asynccnt)
  __builtin_amdgcn_s_wait_asynccnt(0);
#else
  asm volatile("s_wait_asynccnt 0" ::: "memory");
#endif
}

// async 16B copy: global -> LDS (gfx1250 ASYNCcnt path).
// LDS address = low 32 bits of the generic shared-memory pointer (aperture rule).
__device__ __forceinline__ void async_g2l_b128(const float* g, float* l) {
  unsigned ldsa = (unsigned)(uintptr_t)l;
  unsigned long long ga = (unsigned long long)(uintptr_t)g;
  asm volatile("global_load_async_to_lds_b128 %0, %1, off"
               :: "v"(ldsa), "v"(ga) : "memory");
}

// ---------------------------------------------------------------------------
// GEMM with bias: Y[M,Nc] = A[M,K] @ W[K,Nc] + b.  K, Nc compile-time.
// One wave32 computes a 16x16 tile with V_WMMA_F32_16X16X4_F32; 4 waves/block.
// W staged to LDS (zero-padded rows K..Kp-1), B fragments from LDS.
// Requires M % 16 == 0 (100000 = 6250*16).
// ---------------------------------------------------------------------------
template <int K, int Nc>
__global__ __launch_bounds__(128)
void gemm_bias_wmma(const float* __restrict__ A, const float* __restrict__ W,
                    const float* __restrict__ bias, float* __restrict__ Y,
                    int M) {
  constexpr int Kp = (K + 3) & ~3;           // pad K to multiple of 4
  __shared__ float sW[Kp * Nc];

  const int tid = threadIdx.x;
  // async-stage W (K*Nc floats; always divisible by 4 -> 16B chunks)
  static_assert((K * Nc) % 4 == 0, "16B staging");
  for (int idx = tid; idx < (K * Nc) / 4; idx += 128)
    async_g2l_b128(W + idx * 4, &sW[idx * 4]);
  // zero-pad rows K..Kp-1
  for (int idx = tid; idx < (Kp - K) * Nc; idx += 128)
    sW[K * Nc + idx] = 0.0f;
  wait_async0();
  __syncthreads();

  const int lane = tid & 31;
  const int wave = tid >> 5;
  const int m0 = blockIdx.x * 16;
  const int n0 = (blockIdx.y * 4 + wave) * 16;
  const int half = lane >> 4;                // 0: lanes 0-15, 1: lanes 16-31
  const int r    = lane & 15;

  const int row  = m0 + r;                   // A-fragment row (always < M)
  const int col  = n0 + r;                   // B/C/D column
  const int colc = (col < Nc) ? col : (Nc - 1);  // clamp: bad cols never stored

  const float* __restrict__ Arow = A + (size_t)row * K;

  v8f acc = {};
#pragma unroll
  for (int t = 0; t < Kp / 4; ++t) {
    const int ka = 4 * t + 2 * half;         // this lane's K pair
    v2f a, b;
    if (4 * t + 3 < K) {                     // full step: unconditional b64 load
      a.x = Arow[ka];
      a.y = Arow[ka + 1];
    } else {                                 // tail: clamped address + select
      const int i0 = (ka < K) ? ka : 0;
      const int i1 = (ka + 1 < K) ? ka + 1 : 0;
      const float x0 = Arow[i0];
      const float x1 = Arow[i1];
      a.x = (ka < K) ? x0 : 0.0f;
      a.y = (ka + 1 < K) ? x1 : 0.0f;
    }
    // B from LDS: rows >= K are zero-padded -> no selects needed
    b.x = sW[ka * Nc + colc];
    b.y = sW[(ka + 1) * Nc + colc];
    acc = __builtin_amdgcn_wmma_f32_16x16x4_f32(
        /*neg_a=*/false, a, /*neg_b=*/false, b,
        /*c_mod=*/(short)0, acc, /*reuse_a=*/false, /*reuse_b=*/false);
  }

  if (col < Nc) {
    const float bv = bias ? bias[col] : 0.0f;
#pragma unroll
    for (int i = 0; i < 8; ++i) {
      const int rr = m0 + i + 8 * half;      // C/D layout: VGPR i -> M=i (+8 hi)
      Y[(size_t)rr * Nc + col] = acc[i] + bv;
    }
  }
}

// ---------------------------------------------------------------------------
// Utility: fill a float buffer
// ---------------------------------------------------------------------------
__global__ void fill_f32(float* __restrict__ p, float v, size_t n) {
  size_t i = (size_t)blockIdx.x * blockDim.x + threadIdx.x;
  if (i < n) p[i] = v;
}

// atomic float max via signed/unsigned bit trick (init with -inf)
__device__ __forceinline__ void atomicMaxF(float* addr, float val) {
  if (val >= 0.0f) {
    atomicMax((int*)addr, __float_as_int(val));
  } else {
    atomicMin((unsigned int*)addr, __float_as_uint(val));
  }
}

// ---------------------------------------------------------------------------
// Edge pass 1: alpha[e,h] = <q[dst], k[src] + ea[e] @ We> / sqrt(C)
//              amax[dst,h] = max(...).  We (<=9*192 floats) broadcast from LDS.
// ---------------------------------------------------------------------------
__global__ __launch_bounds__(256)
void edge_pass1(const float* __restrict__ q, const float* __restrict__ k,
                const float* __restrict__ ea, const float* __restrict__ We,
                const long long* __restrict__ src, const long long* __restrict__ dst,
                float* __restrict__ alpha, float* __restrict__ amax,
                int E, int H, int C, float inv_sqrt_c) {
  __shared__ float sWe[9 * 192];
  const int hc = H * C;
  for (int idx = threadIdx.x; idx < 9 * hc; idx += 256) sWe[idx] = We[idx];
  __syncthreads();

  const int e = blockIdx.x * 256 + threadIdx.x;
  if (e >= E) return;
  const long long s = src[e], d = dst[e];
  float a9[9];
#pragma unroll
  for (int j = 0; j < 9; ++j) a9[j] = ea[(size_t)e * 9 + j];
  const float* qrow = q + (size_t)d * hc;
  const float* krow = k + (size_t)s * hc;
  __builtin_prefetch(krow, 0, 0);   // global_prefetch_b8
  for (int h = 0; h < H; ++h) {
    float dot = 0.0f;
#pragma unroll 4
    for (int c = 0; c < C; ++c) {
      const int col = h * C + c;
      float ec = 0.0f;
#pragma unroll
      for (int j = 0; j < 9; ++j) ec = fmaf(a9[j], sWe[j * hc + col], ec);
      dot = fmaf(qrow[col], krow[col] + ec, dot);
    }
    const float al = dot * inv_sqrt_c;
    alpha[(size_t)e * H + h] = al;
    atomicMaxF(&amax[(size_t)d * H + h], al);
  }
}

// ---------------------------------------------------------------------------
// Edge pass 2: ex = exp(alpha - amax[dst]); denom[dst,h] += ex;
//              acc[dst,h,c] += ex * (v[src,h,c] + e[h,c])
// (normalization by denom deferred to node kernel: sum(ex*m)/denom)
// ---------------------------------------------------------------------------
__global__ __launch_bounds__(256)
void edge_pass2(const float* __restrict__ alpha, const float* __restrict__ amax,
                const float* __restrict__ v,
                const float* __restrict__ ea, const float* __restrict__ We,
                const long long* __restrict__ src, const long long* __restrict__ dst,
                float* __restrict__ denom, float* __restrict__ acc,
                int E, int H, int C) {
  __shared__ float sWe[9 * 192];
  const int hc = H * C;
  for (int idx = threadIdx.x; idx < 9 * hc; idx += 256) sWe[idx] = We[idx];
  __syncthreads();

  const int e = blockIdx.x * 256 + threadIdx.x;
  if (e >= E) return;
  const long long s = src[e], d = dst[e];
  float a9[9];
#pragma unroll
  for (int j = 0; j < 9; ++j) a9[j] = ea[(size_t)e * 9 + j];
  const float* vrow = v + (size_t)s * hc;
  float* arow = acc + (size_t)d * hc;
  __builtin_prefetch(vrow, 0, 0);
  for (int h = 0; h < H; ++h) {
    const float ex = expf(alpha[(size_t)e * H + h] - amax[(size_t)d * H + h]);
    atomicAdd(&denom[(size_t)d * H + h], ex);
#pragma unroll 4
    for (int c = 0; c < C; ++c) {
      const int col = h * C + c;
      float ec = 0.0f;
#pragma unroll
      for (int j = 0; j < 9; ++j) ec = fmaf(a9[j], sWe[j * hc + col], ec);
      atomicAdd(&arow[col], ex * (vrow[col] + ec));
    }
  }
}

// ---------------------------------------------------------------------------
// Node finish for layers 1 & 2 (H=3, C=64, concat=False):
//   out = relu(mean_h(acc/denom) + skip) [+ prev]
// ---------------------------------------------------------------------------
__global__ __launch_bounds__(256)
void node_finish12(const float* __restrict__ acc, const float* __restrict__ denom,
                   const float* __restrict__ skip, const float* __restrict__ prev,
                   float* __restrict__ out, int N, int H, int C) {
  int idx = blockIdx.x * blockDim.x + threadIdx.x;
  if (idx >= N * C) return;
  const int n = idx / C, c = idx % C;
  float v = 0.0f;
  for (int h = 0; h < H; ++h)
    v += acc[(size_t)n * H * C + h * C + c] / (denom[(size_t)n * H + h] + 1e-16f);
  v = v / (float)H + skip[(size_t)n * C + c];
  v = fmaxf(v, 0.0f);
  if (prev) v += prev[(size_t)n * C + c];
  out[(size_t)n * C + c] = v;
}

// ---------------------------------------------------------------------------
// Node finish for layer 3 (H=1, C=10, concat=True) + log_softmax
// ---------------------------------------------------------------------------
__global__ __launch_bounds__(256)
void node_finish3(const float* __restrict__ acc, const float* __restrict__ denom,
                  const float* __restrict__ skip, float* __restrict__ out, int N) {
  int n = blockIdx.x * blockDim.x + threadIdx.x;
  if (n >= N) return;
  const float dnm = denom[n] + 1e-16f;
  float t[10];
  float mx = -INFINITY;
#pragma unroll
  for (int c = 0; c < 10; ++c) {
    t[c] = acc[(size_t)n * 10 + c] / dnm + skip[(size_t)n * 10 + c];
    mx = fmaxf(mx, t[c]);
  }
  float sum = 0.0f;
#pragma unroll
  for (int c = 0; c < 10; ++c) sum += expf(t[c] - mx);
  const float ls = logf(sum);
#pragma unroll
  for (int c = 0; c < 10; ++c) out[(size_t)n * 10 + c] = t[c] - mx - ls;
}

// ---------------------------------------------------------------------------
// Host orchestration
// ---------------------------------------------------------------------------
template <int K, int Nc>
static inline void launch_gemm(const float* A, const float* W, const float* b,
                               float* Y, int M, hipStream_t st) {
  dim3 grid(M / 16, ((Nc + 15) / 16 + 3) / 4);
  gemm_bias_wmma<K, Nc><<<grid, 128, 0, st>>>(A, W, b, Y, M);
}

static inline void launch_fill(float* p, float v, size_t n, hipStream_t st) {
  fill_f32<<<(unsigned)((n + 255) / 256), 256, 0, st>>>(p, v, n);
}

extern "C" void kernel_launch(void* const* d_in, const int* in_sizes, int n_in,
                              void* d_out, int out_size, void* d_ws, size_t ws_size,
                              hipStream_t stream) {
  const int N = N_NODES, E = N_EDGES;

  const float*     x   = (const float*)d_in[0];
  const long long* ei  = (const long long*)d_in[1];
  const float*     ea  = (const float*)d_in[2];
  const long long* src = ei;
  const long long* dst = ei + E;

  // params order per setup_inputs(): Wq bq Wk bk Wv bv We Wskip bskip
  const float* p1[9]; for (int i = 0; i < 9; ++i) p1[i] = (const float*)d_in[3 + i];
  const float* p2[9]; for (int i = 0; i < 9; ++i) p2[i] = (const float*)d_in[12 + i];
  const float* p3[9]; for (int i = 0; i < 9; ++i) p3[i] = (const float*)d_in[21 + i];

  // Workspace carve-up (floats)
  float* w = (float*)d_ws;
  const size_t Nhc = (size_t)N * 192;
  float* q     = w;                       // N*192
  float* k     = q + Nhc;                 // N*192
  float* v     = k + Nhc;                 // N*192
  float* acc   = v + Nhc;                 // N*192
  float* alpha = acc + Nhc;               // E*3
  float* amax  = alpha + (size_t)E * 3;   // N*3
  float* denom = amax + (size_t)N * 3;    // N*3
  float* skipb = denom + (size_t)N * 3;   // N*64
  float* h1    = skipb + (size_t)N * 64;  // N*64
  float* hbuf  = h1 + (size_t)N * 64;     // N*64

  const int eblocks = E / 256;            // 1600000 = 6250*256 exact
  const float isc64 = 0.125f;             // 1/sqrt(64)
  const float isc10 = 0.31622776601683794f; // 1/sqrt(10)

  // ---------------- Layer 1: din=13, H=3, C=64 ----------------
  launch_gemm<13, 192>(x, p1[0], p1[1], q, N, stream);
  launch_gemm<13, 192>(x, p1[2], p1[3], k, N, stream);
  launch_gemm<13, 192>(x, p1[4], p1[5], v, N, stream);
  launch_gemm<13, 64>(x, p1[7], p1[8], skipb, N, stream);
  launch_fill(amax, -INFINITY, (size_t)N * 3, stream);
  launch_fill(denom, 0.0f, (size_t)N * 3, stream);
  launch_fill(acc, 0.0f, Nhc, stream);
  edge_pass1<<<eblocks, 256, 0, stream>>>(q, k, ea, p1[6], src, dst, alpha, amax,
                                          E, 3, 64, isc64);
  edge_pass2<<<eblocks, 256, 0, stream>>>(alpha, amax, v, ea, p1[6], src, dst,
                                          denom, acc, E, 3, 64);
  node_finish12<<<(N * 64 + 255) / 256, 256, 0, stream>>>(acc, denom, skipb,
                                                          nullptr, h1, N, 3, 64);

  // ---------------- Layer 2: din=64, H=3, C=64 ----------------
  launch_gemm<64, 192>(h1, p2[0], p2[1], q, N, stream);
  launch_gemm<64, 192>(h1, p2[2], p2[3], k, N, stream);
  launch_gemm<64, 192>(h1, p2[4], p2[5], v, N, stream);
  launch_gemm<64, 64>(h1, p2[7], p2[8], skipb, N, stream);
  launch_fill(amax, -INFINITY, (size_t)N * 3, stream);
  launch_fill(denom, 0.0f, (size_t)N * 3, stream);
  launch_fill(acc, 0.0f, Nhc, stream);
  edge_pass1<<<eblocks, 256, 0, stream>>>(q, k, ea, p2[6], src, dst, alpha, amax,
                                          E, 3, 64, isc64);
  edge_pass2<<<eblocks, 256, 0, stream>>>(alpha, amax, v, ea, p2[6], src, dst,
                                          denom, acc, E, 3, 64);
  // hbuf = relu(layer2) + h1   (input to layer 3)
  node_finish12<<<(N * 64 + 255) / 256, 256, 0, stream>>>(acc, denom, skipb,
                                                          h1, hbuf, N, 3, 64);

  // ---------------- Layer 3: din=64, H=1, C=10, concat ----------------
  launch_gemm<64, 10>(hbuf, p3[0], p3[1], q, N, stream);
  launch_gemm<64, 10>(hbuf, p3[2], p3[3], k, N, stream);
  launch_gemm<64, 10>(hbuf, p3[4], p3[5], v, N, stream);
  launch_gemm<64, 10>(hbuf, p3[7], p3[8], skipb, N, stream);
  launch_fill(amax, -INFINITY, (size_t)N, stream);
  launch_fill(denom, 0.0f, (size_t)N, stream);
  launch_fill(acc, 0.0f, (size_t)N * 10, stream);
  edge_pass1<<<eblocks, 256, 0, stream>>>(q, k, ea, p3[6], src, dst, alpha, amax,
                                          E, 1, 10, isc10);
  edge_pass2<<<eblocks, 256, 0, stream>>>(alpha, amax, v, ea, p3[6], src, dst,
                                          denom, acc, E, 1, 10);
  node_finish3<<<(N + 255) / 256, 256, 0, stream>>>(acc, denom, skipb,
                                                    (float*)d_out, N);
  (void)in_sizes; (void)n_in; (void)out_size; (void)ws_size;
}